// PhACE_43516608643287
// MI455X (gfx1250) — compile-verified
//
#include <hip/hip_runtime.h>
#include <math.h>

#define RCUT 5.0f
#define NCH 16
#define KDIM 64
#define MP_SCALE 0.5f
#define MAX_NNZ 4096
#define CG_SIZE (4*4*4*343)

typedef __attribute__((ext_vector_type(2))) float v2f;
typedef __attribute__((ext_vector_type(8))) float v8f;

__device__ __forceinline__ int row_to_l(int r) {
  return (r < 1) ? 0 : (r < 4) ? 1 : (r < 9) ? 2 : 3;
}
__device__ __forceinline__ int cg_idx(int l1, int l2, int L, int a, int b, int c) {
  return (((l1 * 4 + l2) * 4 + L) * 343) + (a * 7 + b) * 7 + c;
}

// ---------------- CG table construction (device, deterministic) ----------------
__device__ double dfact(int n) { double r = 1.0; for (int i = 2; i <= n; ++i) r *= (double)i; return r; }

__device__ double clebsch(int j1, int m1, int j2, int m2, int J, int M) {
  if (m1 + m2 != M || J < abs(j1 - j2) || J > j1 + j2) return 0.0;
  double pref = sqrt((2.0 * J + 1.0) * dfact(J + j1 - j2) * dfact(J - j1 + j2) *
                     dfact(j1 + j2 - J) / dfact(j1 + j2 + J + 1));
  pref *= sqrt(dfact(J + M) * dfact(J - M) * dfact(j1 - m1) * dfact(j1 + m1) *
               dfact(j2 - m2) * dfact(j2 + m2));
  double s = 0.0;
  for (int k = 0; k <= j1 + j2 - J; k++) {
    int d0 = k, d1 = j1 + j2 - J - k, d2 = j1 - m1 - k, d3 = j2 + m2 - k,
        d4 = J - j2 + m1 + k, d5 = J - j1 - m2 + k;
    if (d0 < 0 || d1 < 0 || d2 < 0 || d3 < 0 || d4 < 0 || d5 < 0) continue;
    double den = dfact(d0) * dfact(d1) * dfact(d2) * dfact(d3) * dfact(d4) * dfact(d5);
    s += ((k & 1) ? -1.0 : 1.0) / den;
  }
  return pref * s;
}

__device__ void u_real_mat(int l, double* Ur, double* Ui) {
  int n = 2 * l + 1;
  for (int i = 0; i < n * n; i++) { Ur[i] = 0.0; Ui[i] = 0.0; }
  Ur[l * n + l] = 1.0;
  const double is2 = 0.70710678118654752440;
  for (int m = 1; m <= l; m++) {
    double sgn = (m & 1) ? -1.0 : 1.0;
    Ur[(l + m) * n + (l + m)] = sgn * is2;
    Ur[(l + m) * n + (l - m)] = is2;
    Ui[(l - m) * n + (l - m)] = is2;
    Ui[(l - m) * n + (l + m)] = -sgn * is2;
  }
}

__global__ void init_cg_kernel(float* cg, unsigned int* nnz_idx, float* nnz_val, int* nnz_cnt) {
  int t = threadIdx.x;
  if (t < 64) {
    int l1 = t >> 4, l2 = (t >> 2) & 3, L = t & 3;
    float* base = cg + (((l1 * 4 + l2) * 4 + L) * 343);
    for (int i = 0; i < 343; i++) base[i] = 0.f;
    bool valid = (L >= abs(l1 - l2)) && (L <= l1 + l2);
    if (valid) {
      int n1 = 2 * l1 + 1, n2 = 2 * l2 + 1, nL = 2 * L + 1;
      double Cc[343];
      for (int i = 0; i < n1 * n2 * nL; i++) Cc[i] = 0.0;
      for (int m1 = -l1; m1 <= l1; m1++)
        for (int m2 = -l2; m2 <= l2; m2++) {
          int M = m1 + m2;
          if (M >= -L && M <= L)
            Cc[((l1 + m1) * n2 + (l2 + m2)) * nL + (L + M)] = clebsch(l1, m1, l2, m2, L, M);
        }
      double U1r[49], U1i[49], U2r[49], U2i[49], U3r[49], U3i[49];
      u_real_mat(l1, U1r, U1i);
      u_real_mat(l2, U2r, U2i);
      u_real_mat(L, U3r, U3i);
      double Tr[343], Ti[343], sr = 0.0, si = 0.0;
      for (int a = 0; a < n1; a++)
        for (int b = 0; b < n2; b++)
          for (int c = 0; c < nL; c++) {
            double tr = 0.0, ti = 0.0;
            for (int m = 0; m < n1; m++) {
              double u1r = U1r[a * n1 + m], u1i = U1i[a * n1 + m];
              if (u1r == 0.0 && u1i == 0.0) continue;
              for (int nn = 0; nn < n2; nn++) {
                double u2r = U2r[b * n2 + nn], u2i = U2i[b * n2 + nn];
                if (u2r == 0.0 && u2i == 0.0) continue;
                double pr = u1r * u2r - u1i * u2i;
                double pi = u1r * u2i + u1i * u2r;
                for (int o = 0; o < nL; o++) {
                  double cc = Cc[(m * n2 + nn) * nL + o];
                  if (cc == 0.0) continue;
                  double u3r = U3r[c * nL + o], u3i = -U3i[c * nL + o];  // conj
                  tr += (pr * u3r - pi * u3i) * cc;
                  ti += (pr * u3i + pi * u3r) * cc;
                }
              }
            }
            Tr[(a * n2 + b) * nL + c] = tr;
            Ti[(a * n2 + b) * nL + c] = ti;
            sr += fabs(tr); si += fabs(ti);
          }
      bool use_re = (sr >= si);
      for (int a = 0; a < n1; a++)
        for (int b = 0; b < n2; b++)
          for (int c = 0; c < nL; c++)
            base[(a * 7 + b) * 7 + c] =
                (float)(use_re ? Tr[(a * n2 + b) * nL + c] : Ti[(a * n2 + b) * nL + c]);
    }
  }
  __syncthreads();
  if (t == 0) {
    int cnt = 0;
    for (int l1 = 0; l1 < 4; l1++)
      for (int l2 = 0; l2 < 4; l2++)
        for (int L = 0; L < 4; L++) {
          if (L < abs(l1 - l2) || L > l1 + l2) continue;
          for (int a = 0; a < 2 * l1 + 1; a++)
            for (int b = 0; b < 2 * l2 + 1; b++)
              for (int c = 0; c < 2 * L + 1; c++) {
                float v = cg[cg_idx(l1, l2, L, a, b, c)];
                if (v != 0.f && cnt < MAX_NNZ) {
                  unsigned int outr = (unsigned)(L * L + c);
                  unsigned int in1 = (unsigned)(l1 * l1 + a);
                  unsigned int in2 = (unsigned)(l2 * l2 + b);
                  nnz_idx[cnt] = outr | (in1 << 8) | (in2 << 16);
                  nnz_val[cnt] = v;
                  cnt++;
                }
              }
        }
    *nnz_cnt = cnt;
  }
}

// ---------------- Edge geometry: spherical harmonics + radial basis ----------------
__global__ void edge_geom_kernel(const float* __restrict__ pos,
                                 const int* __restrict__ senders,
                                 const int* __restrict__ receivers,
                                 float* __restrict__ sh, float* __restrict__ basis, int E) {
  int e = blockIdx.x * blockDim.x + threadIdx.x;
  if (e >= E) return;
  int s = senders[e], r = receivers[e];
  float dx = pos[r * 3 + 0] - pos[s * 3 + 0];
  float dy = pos[r * 3 + 1] - pos[s * 3 + 1];
  float dz = pos[r * 3 + 2] - pos[s * 3 + 2];
  float d = sqrtf(dx * dx + dy * dy + dz * dz + 1e-12f);
  float x = dx / d, y = dy / d, z = dz / d;
  float* Y = sh + (size_t)e * 16;
  Y[0] = 0.28209479177387814f;
  const float c1 = 0.4886025119029199f;
  Y[1] = c1 * y; Y[2] = c1 * z; Y[3] = c1 * x;
  Y[4] = 1.0925484305920792f * x * y;
  Y[5] = 1.0925484305920792f * y * z;
  Y[6] = 0.31539156525252005f * (3.f * z * z - 1.f);
  Y[7] = 1.0925484305920792f * x * z;
  Y[8] = 0.5462742152960396f * (x * x - y * y);
  Y[9]  = 0.5900435899266435f * y * (3.f * x * x - y * y);
  Y[10] = 2.890611442640554f * x * y * z;
  Y[11] = 0.4570457994644658f * y * (5.f * z * z - 1.f);
  Y[12] = 0.3731763325901154f * z * (5.f * z * z - 3.f);
  Y[13] = 0.4570457994644658f * x * (5.f * z * z - 1.f);
  Y[14] = 1.445305721320277f * z * (x * x - y * y);
  Y[15] = 0.5900435899266435f * x * (x * x - 3.f * y * y);
  float fc = (d < RCUT) ? 0.5f * (cosf(3.14159265358979323846f * d / RCUT) + 1.f) : 0.f;
  const float sigma = RCUT / 8.f;
  float* B = basis + (size_t)e * 8;
#pragma unroll
  for (int j = 0; j < 8; j++) {
    float mu = (RCUT * j) / 7.f;  // linspace(0, RC, 8)
    float td = (d - mu) / sigma;
    B[j] = expf(-0.5f * td * td) * fc;
  }
}

// ---------------- Phase 1: scatter outer-product messages ----------------
__global__ void phase1_kernel(const float* __restrict__ sh, const float* __restrict__ basis,
                              const float* __restrict__ emb, const int* __restrict__ species,
                              const float* __restrict__ W_inv,
                              const int* __restrict__ senders, const int* __restrict__ receivers,
                              float* __restrict__ fA, int E) {
  long long gid = (long long)blockIdx.x * blockDim.x + threadIdx.x;
  if (gid >= (long long)E * KDIM) return;
  int e = (int)(gid >> 6);
  int k = (int)(gid & 63);
  int nIdx = k >> 4, c = k & 15;
  int s = senders[e], r = receivers[e];
  float xc = emb[species[s] * NCH + c];
  float b[8];
  const float* B = basis + (size_t)e * 8;
#pragma unroll
  for (int j = 0; j < 8; j++) b[j] = B[j];
  float R[4];
#pragma unroll
  for (int l = 0; l < 4; l++) {
    float acc = 0.f;
#pragma unroll
    for (int j = 0; j < 8; j++) acc += b[j] * W_inv[l * 32 + j * 4 + nIdx];
    R[l] = acc;
  }
  const float* Y = sh + (size_t)e * 16;
  float* dst = fA + (size_t)r * 1024 + k;
#pragma unroll
  for (int row = 0; row < 16; ++row) {
    int l = row_to_l(row);
    float v = MP_SCALE * Y[row] * R[l] * xc;
    atomicAdd(dst + row * 64, v);
  }
}

// ---------------- Phase 2: per-atom CG bilinear + embedding gate ----------------
__global__ void phase2_kernel(const float* __restrict__ fA, float* __restrict__ fB,
                              const float* __restrict__ emb, const int* __restrict__ species,
                              const unsigned int* __restrict__ nnz_idx,
                              const float* __restrict__ nnz_val,
                              const int* __restrict__ nnz_cnt, int N) {
  __shared__ unsigned int sidx[MAX_NNZ];
  __shared__ float sval[MAX_NNZ];
  int cnt = *nnz_cnt;
  for (int i = threadIdx.x; i < cnt; i += blockDim.x) { sidx[i] = nnz_idx[i]; sval[i] = nnz_val[i]; }
  __syncthreads();
  long long gid = (long long)blockIdx.x * blockDim.x + threadIdx.x;
  if (gid >= (long long)N * KDIM) return;
  int n = (int)(gid >> 6);
  int k = (int)(gid & 63);
  float f[16], nv[16];
  const float* src = fA + (size_t)n * 1024 + k;
#pragma unroll
  for (int r = 0; r < 16; r++) { f[r] = src[r * 64]; nv[r] = f[r]; }
  for (int i = 0; i < cnt; i++) {
    unsigned int p = sidx[i];
    int outr = p & 0xff, i1 = (p >> 8) & 0xff, i2 = (p >> 16) & 0xff;
    nv[outr] += sval[i] * f[i1] * f[i2];
  }
  float ek = emb[species[n] * NCH + (k & 15)];
  float* dst = fB + (size_t)n * 1024 + k;
#pragma unroll
  for (int r = 0; r < 16; r++) dst[r * 64] = nv[r] * ek;
}

// ---------------- Phase 3: equivariant message passing scatter ----------------
__global__ void phase3_kernel(const float* __restrict__ sh, const float* __restrict__ basis,
                              const float* __restrict__ fB, const float* __restrict__ W_eq,
                              const int* __restrict__ senders, const int* __restrict__ receivers,
                              const unsigned int* __restrict__ nnz_idx,
                              const float* __restrict__ nnz_val,
                              const int* __restrict__ nnz_cnt,
                              float* __restrict__ mp, int E) {
  __shared__ unsigned int sidx[MAX_NNZ];
  __shared__ float sval[MAX_NNZ];
  int cnt = *nnz_cnt;
  for (int i = threadIdx.x; i < cnt; i += blockDim.x) { sidx[i] = nnz_idx[i]; sval[i] = nnz_val[i]; }
  __syncthreads();
  long long gid = (long long)blockIdx.x * blockDim.x + threadIdx.x;
  if (gid >= (long long)E * KDIM) return;
  int e = (int)(gid >> 6);
  int k = (int)(gid & 63);
  int s = senders[e], r = receivers[e];
  float Yv[16], nb[16];
  const float* Y = sh + (size_t)e * 16;
  const float* nbp = fB + (size_t)s * 1024 + k;
#pragma unroll
  for (int i = 0; i < 16; i++) { Yv[i] = Y[i]; nb[i] = nbp[i * 64]; }
  float b[8];
  const float* B = basis + (size_t)e * 8;
#pragma unroll
  for (int j = 0; j < 8; j++) b[j] = B[j];
  float Req[4];
#pragma unroll
  for (int L = 0; L < 4; L++) {
    float acc = 0.f;
#pragma unroll
    for (int j = 0; j < 8; j++) acc += b[j] * W_eq[L * 512 + j * 64 + k];
    Req[L] = acc;
  }
  float m[16];
#pragma unroll
  for (int i = 0; i < 16; i++) m[i] = 0.f;
  for (int i = 0; i < cnt; i++) {
    unsigned int p = sidx[i];
    int outr = p & 0xff, i1 = (p >> 8) & 0xff, i2 = (p >> 16) & 0xff;
    m[outr] += sval[i] * Yv[i1] * nb[i2];
  }
  float* dst = mp + (size_t)r * 1024 + k;
#pragma unroll
  for (int row = 0; row < 16; row++) {
    int L = row_to_l(row);
    float v = MP_SCALE * Req[L] * m[row];
    atomicAdd(dst + row * 64, v);
  }
}

// ---------------- Invariant contraction ----------------
__global__ void invariants_kernel(const float* __restrict__ mp, const float* __restrict__ cg,
                                  float* __restrict__ invb, int N) {
  long long gid = (long long)blockIdx.x * blockDim.x + threadIdx.x;
  if (gid >= (long long)N * KDIM) return;
  int n = (int)(gid >> 6);
  int k = (int)(gid & 63);
  float v[16];
  const float* src = mp + (size_t)n * 1024 + k;
#pragma unroll
  for (int r = 0; r < 16; r++) v[r] = src[r * 64];
  float acc = v[0];
#pragma unroll
  for (int l = 0; l < 4; l++) {
    int nl = 2 * l + 1;
    for (int i = 0; i < nl; i++)
      for (int j = 0; j < nl; j++) {
        float c = cg[cg_idx(l, l, 0, i, j, 0)];
        acc += c * v[l * l + i] * v[l * l + j];
      }
  }
  invb[(size_t)n * 64 + k] = acc;
}

// ---------------- MLP head: WMMA f32 16x16x4 GEMMs + SiLU + reduce ----------------
// All operands staged in LDS: A tile (bounds handled at staging, zero-padded),
// W1/W2 weight matrices (16 KB each). Inner loops are pure ds_load + v_wmma.
__device__ __forceinline__ float silu(float v) { return v / (1.f + expf(-v)); }

__global__ void mlp_kernel(const float* __restrict__ invb, const float* __restrict__ W1,
                           const float* __restrict__ W2, const float* __restrict__ wl,
                           float* __restrict__ out, int N) {
  __shared__ float W1s[64 * 64];
  __shared__ float W2s[64 * 64];
  __shared__ float As[16 * 64];
  __shared__ float hs[16 * 64];
  __shared__ float red[128];
  int tid = threadIdx.x;
  int wid = tid >> 5;    // wave 0..3 -> column tile
  int lane = tid & 31;
  int hi = lane >> 4;    // half-wave -> K-pair select
  int m = lane & 15;     // row (A) / column (B,C,D) index within tile
  int colBase = wid * 16;
  int rowBase = blockIdx.x * 16;

  // Stage weights into LDS (1024 float4 each, 128 threads -> 8 iters)
  for (int i = tid; i < 1024; i += 128) {
    ((float4*)W1s)[i] = ((const float4*)W1)[i];
    ((float4*)W2s)[i] = ((const float4*)W2)[i];
  }
  // Stage inv tile (16 rows x 64 cols = 256 float4), zero-fill padded rows
  for (int i = tid; i < 256; i += 128) {
    int row = i >> 4, c4 = i & 15;
    int grow = rowBase + row;
    float4 v;
    if (grow < N) v = ((const float4*)(invb + (size_t)grow * 64))[c4];
    else { v.x = 0.f; v.y = 0.f; v.z = 0.f; v.w = 0.f; }
    ((float4*)As)[i] = v;
  }
  __syncthreads();

  // h = silu(inv @ W1): 16x16 tile per wave, K=64 via 16 steps of K=4
  v8f acc = {};
#pragma unroll
  for (int k0 = 0; k0 < 64; k0 += 4) {
    int ka = k0 + 2 * hi;
    v2f a = *(const v2f*)(As + m * 64 + ka);
    v2f b;
    b.x = W1s[(ka + 0) * 64 + colBase + m];
    b.y = W1s[(ka + 1) * 64 + colBase + m];
    acc = __builtin_amdgcn_wmma_f32_16x16x4_f32(false, a, false, b, (short)0, acc, false, false);
  }
#pragma unroll
  for (int v = 0; v < 8; ++v) hs[(v + 8 * hi) * 64 + colBase + m] = silu(acc[v]);
  __syncthreads();

  // h2 = silu(h @ W2)
  v8f acc2 = {};
#pragma unroll
  for (int k0 = 0; k0 < 64; k0 += 4) {
    int ka = k0 + 2 * hi;
    v2f a = *(const v2f*)(hs + m * 64 + ka);
    v2f b;
    b.x = W2s[(ka + 0) * 64 + colBase + m];
    b.y = W2s[(ka + 1) * 64 + colBase + m];
    acc2 = __builtin_amdgcn_wmma_f32_16x16x4_f32(false, a, false, b, (short)0, acc2, false, false);
  }

  // e_atom = h2 @ w_last ; global sum
  float wlv = wl[colBase + m];
  float ps = 0.f;
#pragma unroll
  for (int v = 0; v < 8; ++v) ps += silu(acc2[v]) * wlv;
  red[tid] = ps;
  __syncthreads();
  for (int s = 64; s > 0; s >>= 1) {
    if (tid < s) red[tid] += red[tid + s];
    __syncthreads();
  }
  if (tid == 0) atomicAdd(out, red[0]);
}

// ---------------- Host launcher ----------------
extern "C" void kernel_launch(void* const* d_in, const int* in_sizes, int n_in,
                              void* d_out, int out_size, void* d_ws, size_t ws_size,
                              hipStream_t stream) {
  const float* pos    = (const float*)d_in[0];
  const float* emb    = (const float*)d_in[1];
  const float* W_inv  = (const float*)d_in[2];
  const float* W_eq   = (const float*)d_in[3];
  const float* W1     = (const float*)d_in[4];
  const float* W2     = (const float*)d_in[5];
  const float* wl     = (const float*)d_in[6];
  const int* species  = (const int*)d_in[7];
  const int* senders  = (const int*)d_in[8];
  const int* recv     = (const int*)d_in[9];
  int N = in_sizes[0] / 3;
  int E = in_sizes[8];

  char* w = (char*)d_ws;
  size_t off = 0;
  float* cg = (float*)(w + off);                off += (size_t)CG_SIZE * 4;
  unsigned int* nzi = (unsigned int*)(w + off); off += (size_t)MAX_NNZ * 4;
  float* nzv = (float*)(w + off);               off += (size_t)MAX_NNZ * 4;
  int* nzc = (int*)(w + off);                   off += 16;
  float* shb = (float*)(w + off);               off += (size_t)E * 16 * 4;
  float* bas = (float*)(w + off);               off += (size_t)E * 8 * 4;
  float* fA = (float*)(w + off);                off += (size_t)N * 1024 * 4;  // feats, then mp
  float* fB = (float*)(w + off);                off += (size_t)N * 1024 * 4;  // embedded feats
  float* invb = (float*)(w + off);              off += (size_t)N * 64 * 4;

  hipMemsetAsync(d_out, 0, sizeof(float), stream);
  hipMemsetAsync(fA, 0, (size_t)N * 1024 * 4, stream);
  init_cg_kernel<<<1, 64, 0, stream>>>(cg, nzi, nzv, nzc);

  const int bs = 256;
  edge_geom_kernel<<<(E + bs - 1) / bs, bs, 0, stream>>>(pos, senders, recv, shb, bas, E);

  long long te = (long long)E * KDIM;
  long long tn = (long long)N * KDIM;
  phase1_kernel<<<(int)((te + bs - 1) / bs), bs, 0, stream>>>(shb, bas, emb, species, W_inv,
                                                              senders, recv, fA, E);
  phase2_kernel<<<(int)((tn + bs - 1) / bs), bs, 0, stream>>>(fA, fB, emb, species, nzi, nzv, nzc, N);
  hipMemsetAsync(fA, 0, (size_t)N * 1024 * 4, stream);  // reuse fA as mp accumulator
  phase3_kernel<<<(int)((te + bs - 1) / bs), bs, 0, stream>>>(shb, bas, fB, W_eq, senders, recv,
                                                              nzi, nzv, nzc, fA, E);
  invariants_kernel<<<(int)((tn + bs - 1) / bs), bs, 0, stream>>>(fA, cg, invb, N);
  mlp_kernel<<<(N + 15) / 16, 128, 0, stream>>>(invb, W1, W2, wl, (float*)d_out, N);
}